// SingleSourceHandshakingKernel_23347442221224
// MI455X (gfx1250) — compile-verified
//
#include <hip/hip_runtime.h>
#include <math.h>

// ---------------------------------------------------------------------------
// Problem constants (B=4, S=256, H=768)
// ---------------------------------------------------------------------------
#define SEQ     256
#define BATCH   4
#define HID     768
#define N2H     1536                 // 2*H  (g columns then v columns)
#define MROWS   (BATCH * SEQ)        // 1024 rows of X
#define NPAIRS  32896                // S*(S+1)/2

typedef float v2f __attribute__((ext_vector_type(2)));
typedef float v4f __attribute__((ext_vector_type(4)));
typedef float v8f __attribute__((ext_vector_type(8)));

// ---------------------------------------------------------------------------
// Kernel 1: fp32 WMMA GEMM.
//   GV[m][n] = sum_k X[m][k] * W[ n%768 ][ (n/768)*768 + k ]
// (n < 768  -> g column,  n >= 768 -> v column; reduction dim is contiguous
//  in memory for both operands -> pure b64 loads per lane.)
// One wave computes a 16(M) x 64(N) tile: the A fragment is reused across 4
// accumulators (4 x v_wmma_f32_16x16x4_f32 per K-step of 4).
// ---------------------------------------------------------------------------
__global__ __launch_bounds__(32) void gv_gemm_wmma(
    const float* __restrict__ X,     // [1024][768]
    const float* __restrict__ W,     // [768][1536]
    float* __restrict__ GV)          // [1024][1536]
{
  const int lane  = threadIdx.x;     // 0..31 (wave32)
  const int laneM = lane & 15;       // M row / N col within tile
  const int halfK = lane >> 4;       // 0: K=0,1   1: K=2,3   (ISA A/B layout)

  const int m0   = blockIdx.x << 4;  // 16 output rows
  const int n0   = blockIdx.y << 6;  // 64 output cols (never straddles 768)
  const int half = n0 / HID;         // 0 => g half of W, 1 => v half
  const int o0   = n0 % HID;         // W row for first column of tile

  const float* __restrict__ arow = X + (size_t)(m0 + laneM) * HID;

  const float* bp0 = W + (size_t)(o0 + 0 * 16 + laneM) * N2H + half * HID;
  const float* bp1 = W + (size_t)(o0 + 1 * 16 + laneM) * N2H + half * HID;
  const float* bp2 = W + (size_t)(o0 + 2 * 16 + laneM) * N2H + half * HID;
  const float* bp3 = W + (size_t)(o0 + 3 * 16 + laneM) * N2H + half * HID;

  v8f acc0 = {}, acc1 = {}, acc2 = {}, acc3 = {};

#pragma unroll 4
  for (int k0 = 0; k0 < HID; k0 += 4) {
    const int ka = k0 + 2 * halfK;

    v2f a;                       // A 16x4 fp32 fragment (2 VGPRs/lane)
    a.x = arow[ka];
    a.y = arow[ka + 1];

    v2f b0; b0.x = bp0[ka]; b0.y = bp0[ka + 1];
    v2f b1; b1.x = bp1[ka]; b1.y = bp1[ka + 1];
    v2f b2; b2.x = bp2[ka]; b2.y = bp2[ka + 1];
    v2f b3; b3.x = bp3[ka]; b3.y = bp3[ka + 1];

    // (neg_a, A, neg_b, B, c_mod, C, reuse_a, reuse_b)
    acc0 = __builtin_amdgcn_wmma_f32_16x16x4_f32(false, a, false, b0,
                                                 (short)0, acc0, false, false);
    acc1 = __builtin_amdgcn_wmma_f32_16x16x4_f32(false, a, false, b1,
                                                 (short)0, acc1, false, false);
    acc2 = __builtin_amdgcn_wmma_f32_16x16x4_f32(false, a, false, b2,
                                                 (short)0, acc2, false, false);
    acc3 = __builtin_amdgcn_wmma_f32_16x16x4_f32(false, a, false, b3,
                                                 (short)0, acc3, false, false);
  }

  // C/D layout: VGPR r -> (M=r, N=lane) for lanes 0-15, (M=r+8, N=lane-16)
  // for lanes 16-31.
#pragma unroll
  for (int r = 0; r < 8; ++r) {
    const size_t row = (size_t)(m0 + r + 8 * halfK) * N2H;
    GV[row + n0 +  0 + laneM] = acc0[r];
    GV[row + n0 + 16 + laneM] = acc1[r];
    GV[row + n0 + 32 + laneM] = acc2[r];
    GV[row + n0 + 48 + laneM] = acc3[r];
  }
}

// ---------------------------------------------------------------------------
// Kernel 2: pairwise tanh epilogue.  One block per pair p (32896 blocks).
//   out[b, p, :] = tanh( g[b, i(p), :] + v[b, j(p), :] + bias )
// g/v rows live in the 6.3 MB GV workspace (L2-resident); the 404 MB output
// stream uses non-temporal 128-bit stores so it doesn't evict GV from L2.
// ---------------------------------------------------------------------------
#define CUMROW(ii) ((ii) * SEQ - ((ii) * ((ii)-1)) / 2)   // pairs before row ii

__global__ __launch_bounds__(256) void pair_tanh_kernel(
    const float* __restrict__ GV,     // [1024][1536]
    const float* __restrict__ bias,   // [768]
    float* __restrict__ out)          // [4][32896][768]
{
  const int p = blockIdx.x;

  // Invert triangular row index: largest i with CUMROW(i) <= p.
  const int disc = (2 * SEQ + 1) * (2 * SEQ + 1) - 8 * p;  // exact in fp32
  int i = (int)((513.0f - sqrtf((float)disc)) * 0.5f);
  if (i < 0) i = 0;
  if (i > SEQ - 1) i = SEQ - 1;
  while (i > 0 && CUMROW(i) > p) --i;
  while (i < SEQ - 1 && CUMROW(i + 1) <= p) ++i;
  const int j = i + (p - CUMROW(i));

  // BATCH * HID floats per block = 768 x float4 -> 3 iterations @ 256 threads.
  for (int t = threadIdx.x; t < BATCH * (HID / 4); t += 256) {
    const int b  = t / (HID / 4);
    const int h4 = t % (HID / 4);

    const v4f g  = ((const v4f*)(GV + (size_t)(b * SEQ + i) * N2H))[h4];
    const v4f v  = ((const v4f*)(GV + (size_t)(b * SEQ + j) * N2H + HID))[h4];
    const v4f bb = ((const v4f*)bias)[h4];

    v4f o;
    o.x = tanhf(g.x + v.x + bb.x);
    o.y = tanhf(g.y + v.y + bb.y);
    o.z = tanhf(g.z + v.z + bb.z);
    o.w = tanhf(g.w + v.w + bb.w);

    v4f* dst = (v4f*)(out + ((size_t)b * NPAIRS + p) * HID) + h4;
    __builtin_nontemporal_store(o, dst);   // TH=NT on the 404 MB write stream
  }
}

// ---------------------------------------------------------------------------
// Host entry
// ---------------------------------------------------------------------------
extern "C" void kernel_launch(void* const* d_in, const int* in_sizes, int n_in,
                              void* d_out, int out_size, void* d_ws,
                              size_t ws_size, hipStream_t stream) {
  const float* X    = (const float*)d_in[0];   // (4,256,768)
  const float* W    = (const float*)d_in[1];   // (768,1536)
  const float* bias = (const float*)d_in[2];   // (768,)
  float* out = (float*)d_out;
  float* GV  = (float*)d_ws;                   // 1024*1536 fp32 = 6.29 MB

  if (ws_size < (size_t)MROWS * N2H * sizeof(float)) return;

  dim3 gemm_grid(MROWS / 16, N2H / 64);        // 64 x 24 waves
  hipLaunchKernelGGL(gv_gemm_wmma, gemm_grid, dim3(32), 0, stream, X, W, GV);

  hipLaunchKernelGGL(pair_tanh_kernel, dim3(NPAIRS), dim3(256), 0, stream,
                     GV, bias, out);
}